// SparseAudioModel_63763084476977
// MI455X (gfx1250) — compile-verified
//
#include <hip/hip_runtime.h>
#include <stdint.h>

// Problem constants (from reference)
#define B      8
#define NS     32768          // N_SAMPLES
#define NA     256            // N_ATOMS
#define ASZ    1024           // ATOM_SIZE
#define KTOP   256            // TO_KEEP
#define TOT    (NA * NS)      // 8,388,608 elements of x per batch
#define BINS   4096           // 12-bit key histogram
#define CAP    8192           // candidate cap per batch
#define ACCN   (NS + ASZ)     // 33792-float LDS accumulator

typedef __attribute__((ext_vector_type(2))) float v2f;
typedef __attribute__((ext_vector_type(8))) float v8f;

// K5 LDS layout (floats): acc[ACCN] | stage[16*ASZ] | v[16] | spos[16]
#define STAGE_OFF  ACCN                     // 33792
#define VBUF_OFF   (ACCN + 16 * ASZ)        // 50176
#define SPOS_OFF   (VBUF_OFF + 16)          // 50192
#define K5_FLOATS  (SPOS_OFF + 16)          // 50208 -> 200,832 bytes

// ---- order-preserving float <-> uint key --------------------------------
__device__ __forceinline__ uint32_t fkey(float f) {
  uint32_t u = __float_as_uint(f);
  return u ^ ((uint32_t)((int32_t)u >> 31) | 0x80000000u);
}
__device__ __forceinline__ uint32_t fkey_inv(uint32_t k) {
  return k ^ ((uint32_t)((int32_t)(~k) >> 31) | 0x80000000u);
}

// ---- K0: zero the workspace state we depend on --------------------------
__global__ void k0_init(uint32_t* __restrict__ hist,
                        uint32_t* __restrict__ cand_count,
                        uint32_t* __restrict__ finalbuf) {
  int i = blockIdx.x * blockDim.x + threadIdx.x;
  if (i < B * BINS) hist[i] = 0u;
  if (i < B) cand_count[i] = 0u;
  if (i < B * KTOP * 2) finalbuf[i] = 0u;
}

// ---- K1: 4096-bin key histogram per batch (LDS hist, flush w/ atomics) --
__global__ void k1_hist(const float* __restrict__ x, uint32_t* __restrict__ hist) {
  __shared__ uint32_t h[BINS];
  const int tid = threadIdx.x;
  const int b = blockIdx.y;
  for (int i = tid; i < BINS; i += 256) h[i] = 0u;
  __syncthreads();
  const size_t base = (size_t)b * TOT + (size_t)blockIdx.x * 65536u;
  const float4* p = (const float4*)(x + base);
  for (int i = tid; i < 16384; i += 256) {
    float4 f = p[i];
    atomicAdd(&h[fkey(f.x) >> 20], 1u);
    atomicAdd(&h[fkey(f.y) >> 20], 1u);
    atomicAdd(&h[fkey(f.z) >> 20], 1u);
    atomicAdd(&h[fkey(f.w) >> 20], 1u);
  }
  __syncthreads();
  uint32_t* gh = hist + (size_t)b * BINS;
  for (int i = tid; i < BINS; i += 256) {
    uint32_t c = h[i];
    if (c) atomicAdd(&gh[i], c);
  }
}

// ---- K2: find threshold bucket b* per batch -----------------------------
__global__ void k2_thresh(const uint32_t* __restrict__ hist, uint32_t* __restrict__ bstar) {
  const int b = blockIdx.x;
  if (threadIdx.x == 0) {
    const uint32_t* gh = hist + (size_t)b * BINS;
    uint32_t acc = 0u;
    int bin = BINS - 1;
    for (; bin > 0; --bin) {
      uint32_t c = gh[bin];
      if (acc + c >= KTOP) break;
      acc += c;
    }
    bstar[b] = (uint32_t)bin;
  }
}

// ---- K3: compact candidates (bucket >= b*) ------------------------------
__global__ void k3_select(const float* __restrict__ x, const uint32_t* __restrict__ bstar,
                          uint32_t* __restrict__ cand_count, uint32_t* __restrict__ cand) {
  const int tid = threadIdx.x;
  const int b = blockIdx.y;
  const uint32_t thr = bstar[b];
  const uint32_t base = (uint32_t)blockIdx.x * 65536u;
  const float4* p = (const float4*)(x + (size_t)b * TOT + base);
  uint32_t* cc = cand_count + b;
  uint32_t* cb = cand + (size_t)b * CAP * 2;
  for (int i = tid; i < 16384; i += 256) {
    float4 f = p[i];
    float v4[4] = {f.x, f.y, f.z, f.w};
#pragma unroll
    for (int j = 0; j < 4; ++j) {
      uint32_t u = fkey(v4[j]);
      if ((u >> 20) >= thr) {
        uint32_t slot = atomicAdd(cc, 1u);
        if (slot < CAP) {
          cb[slot * 2 + 0] = u;                                      // key
          cb[slot * 2 + 1] = base + (uint32_t)i * 4u + (uint32_t)j;  // flat idx
        }
      }
    }
  }
}

// ---- K4: exact deterministic rank -> top-256 (value,index) in order -----
__global__ void k4_rank(const uint32_t* __restrict__ cand, const uint32_t* __restrict__ cand_count,
                        uint32_t* __restrict__ finalbuf) {
  extern __shared__ uint32_t s[];   // skey[CAP], sidx[CAP]  (64 KB dynamic)
  uint32_t* skey = s;
  uint32_t* sidx = s + CAP;
  const int b = blockIdx.x;
  const int tid = threadIdx.x;
  uint32_t C = cand_count[b];
  if (C > CAP) C = CAP;
  const uint32_t* cb = cand + (size_t)b * CAP * 2;
  for (uint32_t i = tid; i < C; i += 256) {
    skey[i] = cb[i * 2 + 0];
    sidx[i] = cb[i * 2 + 1];
  }
  __syncthreads();
  uint32_t* fb = finalbuf + (size_t)b * KTOP * 2;
  for (uint32_t i = tid; i < C; i += 256) {
    const uint32_t ki = skey[i], di = sidx[i];
    uint32_t r = 0;
    for (uint32_t j = 0; j < C; ++j) {
      uint32_t kj = skey[j];
      r += (uint32_t)((kj > ki) || (kj == ki && sidx[j] < di));
    }
    if (r < KTOP) {
      fb[r * 2 + 0] = fkey_inv(ki);   // original float bits
      fb[r * 2 + 1] = di;
    }
  }
}

// ---- K5: WMMA diag-GEMM + LDS-resident overlapping scatter-add ----------
// One block per batch, 8 waves. For each group of 16 selections:
//   Contrib(16x1024) = diag(v0..v15) x AtomRows(16x1024)
// computed as 4 chained V_WMMA_F32_16X16X4_F32 per 16x16 tile. Atom rows
// are staged into LDS with CDNA5 async global->LDS copies; tile results are
// scattered into a 132KB LDS accumulator with ds_add_f32 at spos[row]+col.
__global__ void k5_scatter(const float* __restrict__ atoms, const uint32_t* __restrict__ finalbuf,
                           float* __restrict__ out) {
  extern __shared__ __align__(16) float smem[];
  const int b = blockIdx.x;
  const int tid = threadIdx.x;
  const int wave = tid >> 5;
  const int lane = tid & 31;
  const int half = lane >> 4;     // 0: lanes 0-15, 1: lanes 16-31
  const int nn = lane & 15;       // N coordinate

  float* stage = smem + STAGE_OFF;
  float* vbuf = smem + VBUF_OFF;
  uint32_t* sposu = (uint32_t*)(smem + SPOS_OFF);
  const uint32_t stage_lds = (uint32_t)(uintptr_t)stage;   // LDS byte address
  const uint32_t* fb = finalbuf + (size_t)b * KTOP * 2;

  for (int i = tid; i < ACCN; i += 256) smem[i] = 0.0f;
  __syncthreads();

  for (int g = 0; g < 16; ++g) {
    // group metadata: value + sample position for the 16 rows
    if (tid < 16) {
      const uint32_t vb = fb[(g * 16 + tid) * 2 + 0];
      const uint32_t id = fb[(g * 16 + tid) * 2 + 1];
      vbuf[tid] = __uint_as_float(vb);
      sposu[tid] = id & (NS - 1);
    }
    // async-stage the 16 atom rows (64 KB): wave w copies rows 2w, 2w+1
#pragma unroll
    for (int rr = 0; rr < 2; ++rr) {
      const int row = wave * 2 + rr;
      const uint32_t id = fb[(g * 16 + row) * 2 + 1];
      const uint32_t aidx = id >> 15;                       // / N_SAMPLES
      const unsigned long long src =
          (unsigned long long)(uintptr_t)(atoms + (size_t)aidx * ASZ);
      const uint32_t ldsrow = stage_lds + (uint32_t)row * (ASZ * 4);
#pragma unroll
      for (int j = 0; j < 8; ++j) {
        const uint32_t doff = (uint32_t)((j * 32 + lane) * 16);
        asm volatile("global_load_async_to_lds_b128 %0, %1, off"
                     :: "v"(ldsrow + doff), "v"(src + doff)
                     : "memory");
      }
    }
    asm volatile("s_wait_asynccnt 0" ::: "memory");
    __syncthreads();   // stage + metadata visible to all waves

    // diag A fragments (16x4 f32 layout: VGPR0 holds K=0|K=2, VGPR1 K=1|K=3
    // across lane halves). A[m,k] = (m == k) ? v[m] : 0 for this K-chunk.
    const int m = nn;            // A/C row for this lane half
    const float vm = vbuf[m];
    v2f afrag[4];
#pragma unroll
    for (int c = 0; c < 4; ++c) {
      const int k0 = c * 4 + (half ? 2 : 0);   // K index of VGPR0 element
      const int k1 = c * 4 + (half ? 3 : 1);   // K index of VGPR1 element
      afrag[c].x = (m == k0) ? vm : 0.0f;
      afrag[c].y = (m == k1) ? vm : 0.0f;
    }
    // per-lane row shifts for the C/D layout (VGPR r -> M = r + 8*half)
    uint32_t srow[8];
#pragma unroll
    for (int r = 0; r < 8; ++r) srow[r] = sposu[r + 8 * half];

    // each wave owns 8 of the 64 column tiles
#pragma unroll 1
    for (int tt = 0; tt < 8; ++tt) {
      const int colbase = (wave * 8 + tt) * 16;
      v8f d = {0.f, 0.f, 0.f, 0.f, 0.f, 0.f, 0.f, 0.f};
#pragma unroll
      for (int c = 0; c < 4; ++c) {
        // B fragment (4x16 f32): VGPRq holds K=q (lanes 0-15) / K=q+2 (16-31)
        v2f bfrag;
        bfrag.x = stage[(c * 4 + (half ? 2 : 0)) * ASZ + colbase + nn];
        bfrag.y = stage[(c * 4 + (half ? 3 : 1)) * ASZ + colbase + nn];
        d = __builtin_amdgcn_wmma_f32_16x16x4_f32(
                false, afrag[c], false, bfrag, (short)0, d, false, false);
      }
      // scatter the 16x16 tile: lane holds (M = r + 8*half, N = nn)
#pragma unroll
      for (int r = 0; r < 8; ++r) {
        atomicAdd(&smem[srow[r] + (uint32_t)(colbase + nn)], d[r]);
      }
    }
    __syncthreads();   // done reading stage before next group overwrites it
  }

  float4* dst = (float4*)(out + (size_t)b * NS);
  const float4* sa = (const float4*)smem;
  for (int i = tid; i < NS / 4; i += 256) dst[i] = sa[i];
}

// ---- host launcher ------------------------------------------------------
extern "C" void kernel_launch(void* const* d_in, const int* in_sizes, int n_in,
                              void* d_out, int out_size, void* d_ws, size_t ws_size,
                              hipStream_t stream) {
  const float* x     = (const float*)d_in[0];   // [8, 256*32768]
  const float* atoms = (const float*)d_in[1];   // [1, 256, 1024]
  float* out = (float*)d_out;                   // [8, 1, 32768]

  // workspace layout (bytes): hist | bstar | cand_count | cand | final
  char* ws = (char*)d_ws;
  uint32_t* hist       = (uint32_t*)(ws);                              // 131072 B
  uint32_t* bstar      = (uint32_t*)(ws + 131072);                     //     32 B
  uint32_t* cand_count = (uint32_t*)(ws + 131072 + 64);                //     32 B
  uint32_t* cand       = (uint32_t*)(ws + 131072 + 128);               // 524288 B
  uint32_t* finalbuf   = (uint32_t*)(ws + 131072 + 128 + 524288);      //  16384 B

  // allow >64KB dynamic LDS (gfx1250: 320KB/WG)
  (void)hipFuncSetAttribute((const void*)k5_scatter,
                            hipFuncAttributeMaxDynamicSharedMemorySize,
                            (int)(K5_FLOATS * sizeof(float)));
  (void)hipFuncSetAttribute((const void*)k4_rank,
                            hipFuncAttributeMaxDynamicSharedMemorySize,
                            (int)(2 * CAP * sizeof(uint32_t)));

  k0_init  <<<dim3(128),    dim3(256), 0, stream>>>(hist, cand_count, finalbuf);
  k1_hist  <<<dim3(128, B), dim3(256), 0, stream>>>(x, hist);
  k2_thresh<<<dim3(B),      dim3(32),  0, stream>>>(hist, bstar);
  k3_select<<<dim3(128, B), dim3(256), 0, stream>>>(x, bstar, cand_count, cand);
  k4_rank  <<<dim3(B),      dim3(256), 2 * CAP * sizeof(uint32_t), stream>>>(cand, cand_count, finalbuf);
  k5_scatter<<<dim3(B),     dim3(256), K5_FLOATS * sizeof(float), stream>>>(atoms, finalbuf, out);
}